// ReEig_39161511805475
// MI455X (gfx1250) — compile-verified
//
#include <hip/hip_runtime.h>

typedef __attribute__((ext_vector_type(2))) float v2f;
typedef __attribute__((ext_vector_type(8))) float v8f;

#define C_DIM     128
#define PITCH     132          // row pitch in floats: bank-conflict-free + 16B aligned rows
#define NS_ITERS  26
#define EPS_CLAMP 1e-4f
#define EPS_PERT  1e-6f
#define THREADS   256          // 8 wave32 per workgroup

// Compute a 16x128 strip of D = A * B, where B is SYMMETRIC so its
// 4x16 WMMA fragments can be fetched with row-major addressing
// (B_op(k,n) = B[n][k] = B[k][n]). A, B, D are PITCH-row LDS matrices.
// Wave `wave` produces output rows [16*wave, 16*wave+16).
__device__ __forceinline__ void mm_sym_strip(const float* __restrict__ A,
                                             const float* __restrict__ Bsym,
                                             float* __restrict__ D,
                                             int wave, int lane) {
    const int l  = lane & 15;        // M (for A) / N (for B) index within tile
    const int kh = (lane >> 4) << 1; // K offset contributed by lane half
    const float* arow = A + (wave * 16 + l) * PITCH + kh;

    v8f acc[8] = {};
    for (int k0 = 0; k0 < C_DIM; k0 += 4) {
        v2f a = *(const v2f*)(arow + k0);
#pragma unroll
        for (int tn = 0; tn < 8; ++tn) {
            v2f b = *(const v2f*)(Bsym + (tn * 16 + l) * PITCH + k0 + kh);
            acc[tn] = __builtin_amdgcn_wmma_f32_16x16x4_f32(
                false, a, false, b, (short)0, acc[tn], false, false);
        }
    }
    // C/D layout: VGPR r, lane l+16h -> element (M = r + 8h, N = l)
    const int rbase = wave * 16 + ((lane >> 4) << 3);
#pragma unroll
    for (int tn = 0; tn < 8; ++tn) {
        const int col = tn * 16 + l;
#pragma unroll
        for (int r = 0; r < 8; ++r) {
            D[(rbase + r) * PITCH + col] = acc[tn][r];
        }
    }
}

__global__ __launch_bounds__(THREADS, 1)
void reeig_ns_kernel(const float* __restrict__ x, float* __restrict__ out) {
    extern __shared__ float smem[];
    float* Ms = smem;                         // M = x + diag(pert) - eps*I
    float* Xs = Ms + C_DIM * PITCH;           // Newton-Schulz iterate -> sign(M)
    float* Ts = Xs + C_DIM * PITCH;           // X*X  (also reduction scratch / final M*S)
    float* Ws = Ts + C_DIM * PITCH;           // X*(X*X)

    const int tid  = threadIdx.x;
    const int lane = tid & 31;
    const int wave = tid >> 5;
    const long long base = (long long)blockIdx.x * (C_DIM * C_DIM);

    // ---- Load M into LDS (vectorized), apply diagonal shift, accumulate ||M||_F^2
    float ss = 0.0f;
    for (int c = tid; c < (C_DIM * C_DIM) / 4; c += THREADS) {
        const int row  = c >> 5;          // 32 float4 chunks per row
        const int col4 = (c & 31) << 2;
        float4 v = ((const float4*)(x + base))[c];
        float vv[4] = {v.x, v.y, v.z, v.w};
#pragma unroll
        for (int u = 0; u < 4; ++u) {
            if (col4 + u == row) vv[u] += EPS_PERT * (float)row - EPS_CLAMP;
            ss += vv[u] * vv[u];
        }
        float4 w4 = {vv[0], vv[1], vv[2], vv[3]};
        *(float4*)(Ms + row * PITCH + col4) = w4;   // 16B-aligned (PITCH*4 % 16 == 0)
    }

    // ---- Block reduction of ss (Ts as scratch), s = 1/||M||_F
    Ts[tid] = ss;
    __syncthreads();
    for (int off = THREADS >> 1; off > 0; off >>= 1) {
        if (tid < off) Ts[tid] += Ts[tid + off];
        __syncthreads();
    }
    const float s = rsqrtf(Ts[0] + 1e-30f);
    __syncthreads();

    // ---- X0 = s * M   (spectral norm <= 1: Newton-Schulz converges to sign(M))
    for (int e = tid; e < C_DIM * C_DIM; e += THREADS) {
        const int i = e >> 7, j = e & 127;
        Xs[i * PITCH + j] = Ms[i * PITCH + j] * s;
    }
    __syncthreads();

    // ---- Newton-Schulz: X <- 1.5 X - 0.5 X (X X), with re-symmetrization
    for (int it = 0; it < NS_ITERS; ++it) {
        mm_sym_strip(Xs, Xs, Ts, wave, lane);     // T = X * X   (symmetric)
        __syncthreads();
        mm_sym_strip(Xs, Ts, Ws, wave, lane);     // W = X * T
        __syncthreads();
        for (int e = tid; e < C_DIM * C_DIM; e += THREADS) {
            const int i = e >> 7, j = e & 127;
            const float xv = Xs[i * PITCH + j];
            const float wv = Ws[i * PITCH + j] + Ws[j * PITCH + i]; // sym(W)*2
            Xs[i * PITCH + j] = 1.5f * xv - 0.25f * wv;             // stays symmetric
        }
        __syncthreads();
    }

    // ---- F = M * sign(M) = |M|   (into Ts)
    mm_sym_strip(Ms, Xs, Ts, wave, lane);
    __syncthreads();

    // ---- out = 0.5*M + eps*I + 0.5*sym(F)
    for (int e = tid; e < C_DIM * C_DIM; e += THREADS) {
        const int i = e >> 7, j = e & 127;
        float val = 0.5f * Ms[i * PITCH + j]
                  + 0.25f * (Ts[i * PITCH + j] + Ts[j * PITCH + i]);
        if (i == j) val += EPS_CLAMP;
        out[base + e] = val;
    }
}

extern "C" void kernel_launch(void* const* d_in, const int* in_sizes, int n_in,
                              void* d_out, int out_size, void* d_ws, size_t ws_size,
                              hipStream_t stream) {
    (void)n_in; (void)out_size; (void)d_ws; (void)ws_size;
    const float* x = (const float*)d_in[0];
    float* out = (float*)d_out;
    const int B = in_sizes[0] / (C_DIM * C_DIM);   // 4096

    const size_t shmem = (size_t)4 * C_DIM * PITCH * sizeof(float);  // 270336 B
    (void)hipFuncSetAttribute((const void*)reeig_ns_kernel,
                              hipFuncAttributeMaxDynamicSharedMemorySize,
                              (int)shmem);
    reeig_ns_kernel<<<B, THREADS, shmem, stream>>>(x, out);
}